// MultiHeadAttention_2370821948096
// MI455X (gfx1250) — compile-verified
//
#include <hip/hip_runtime.h>
#include <hip/hip_bf16.h>

// MHA forward for B=2, L=2048, D=1024, H=16, HD=64 (fp32).
// Matrix math: V_WMMA_F32_16X16X4_F32 (full-precision fp32 WMMA path).
// Data movement: GLOBAL_LOAD_ASYNC_TO_LDS_B128 (ASYNCcnt) with double-buffered
// LDS tiles so the async engine fills tile t+1 while WMMAs consume tile t.

typedef __attribute__((ext_vector_type(2))) float v2f;
typedef __attribute__((ext_vector_type(8))) float v8f;

#define WMMA_F32(a, b, c) \
  __builtin_amdgcn_wmma_f32_16x16x4_f32(false, (a), false, (b), (short)0, (c), false, false)

// Async 16B copy global -> LDS. LDS operand is the LDS byte offset in a VGPR
// (low 32 bits of a flat shared pointer ARE the LDS offset per the aperture
// mapping). Tracked by ASYNCcnt.
__device__ __forceinline__ void async_copy16(const float* lds_dst, const float* gsrc) {
  uint32_t loff = (uint32_t)(uintptr_t)lds_dst;
  uint64_t ga   = (uint64_t)(uintptr_t)gsrc;
  asm volatile("global_load_async_to_lds_b128 %0, %1, off"
               :: "v"(loff), "v"(ga) : "memory");
}
__device__ __forceinline__ void wait_async0() {
  asm volatile("s_wait_asynccnt 0" ::: "memory");
}

// ---------------------------------------------------------------------------
// Tiled fp32 GEMM: C[M,N] = A[M,K] * B[K,N], row-major, M%128==0, N%64==0,
// K%16==0. Block tile 128x64, 8 waves, each wave owns 32x32 (2x2 WMMA accs).
// KT=16 double-buffered through LDS via async-to-LDS copies.
// ---------------------------------------------------------------------------
__global__ __launch_bounds__(256) void gemm_f32_kernel(
    const float* __restrict__ A, const float* __restrict__ Bm,
    float* __restrict__ C, int M, int N, int K) {
  __shared__ __attribute__((aligned(16))) float As[2][128 * 20];  // padded rows: bank-safe + b64-aligned
  __shared__ __attribute__((aligned(16))) float Bs[2][16 * 64];

  const int tid  = threadIdx.x;
  const int lane = tid & 31;
  const int wid  = tid >> 5;
  const int half = lane >> 4;       // 0: lanes 0-15, 1: lanes 16-31
  const int l16  = lane & 15;
  const int wm   = (wid & 3) * 32;  // wave M offset inside block tile
  const int wn   = (wid >> 2) * 32; // wave N offset inside block tile
  const int m0   = blockIdx.y * 128;
  const int n0   = blockIdx.x * 64;

  // Per-thread staging slots: A tile 128x16 = 512 float4 (2/thread),
  // B tile 16x64 = 256 float4 (1/thread).
  const int a0row = (tid) >> 2,         a0cv = ((tid) & 3) * 4;
  const int a1row = (tid + 256) >> 2,   a1cv = ((tid + 256) & 3) * 4;
  const int brow  = tid >> 4,           bcv  = (tid & 15) * 4;

  auto issue_tile = [&](int buf, int k0) {
    async_copy16(&As[buf][a0row * 20 + a0cv], A + (size_t)(m0 + a0row) * K + k0 + a0cv);
    async_copy16(&As[buf][a1row * 20 + a1cv], A + (size_t)(m0 + a1row) * K + k0 + a1cv);
    async_copy16(&Bs[buf][brow * 64 + bcv],   Bm + (size_t)(k0 + brow) * N + n0 + bcv);
  };

  v8f acc[2][2] = {};
  const int nkt = K >> 4;
  issue_tile(0, 0);

  for (int kt = 0; kt < nkt; ++kt) {
    const int cur = kt & 1;
    wait_async0();        // my async fills of As/Bs[cur] are done
    __syncthreads();      // everyone's fills done; prev reads of [cur^1] done
    if (kt + 1 < nkt) issue_tile(cur ^ 1, (kt + 1) * 16);

    const float* as = As[cur];
    const float* bs = Bs[cur];
#pragma unroll
    for (int kk = 0; kk < 16; kk += 4) {
      v2f af[2], bf[2];
#pragma unroll
      for (int i = 0; i < 2; ++i) {
        // A frag (16x4): lane row = l16, K = kk + 2*half + {0,1}
        af[i] = *(const v2f*)(&as[(wm + i * 16 + l16) * 20 + kk + 2 * half]);
        // B frag (4x16): lane col = l16, rows kk + 2*half + {0,1}
        bf[i].x = bs[(kk + 2 * half) * 64 + wn + i * 16 + l16];
        bf[i].y = bs[(kk + 2 * half + 1) * 64 + wn + i * 16 + l16];
      }
#pragma unroll
      for (int i = 0; i < 2; ++i)
#pragma unroll
        for (int j = 0; j < 2; ++j)
          acc[i][j] = WMMA_F32(af[i], bf[j], acc[i][j]);
    }
    __syncthreads();      // reads of [cur] done before it is refilled at kt+2
  }

  // C/D layout: VGPR r: lanes 0-15 -> row r, lanes 16-31 -> row r+8; col = l16.
#pragma unroll
  for (int i = 0; i < 2; ++i)
#pragma unroll
    for (int j = 0; j < 2; ++j)
#pragma unroll
      for (int r = 0; r < 8; ++r) {
        int row = m0 + wm + i * 16 + r + 8 * half;
        int col = n0 + wn + j * 16 + l16;
        C[(size_t)row * N + col] = acc[i][j][r];
      }
}

// ---------------------------------------------------------------------------
// RoPE applied in place to q and k thirds of qkv[4096][3072].
// One thread per rotated pair: 4096 tokens * 2 (q,k) * 16 heads * 32 pairs.
// ---------------------------------------------------------------------------
__global__ __launch_bounds__(256) void rope_kernel(
    float* __restrict__ qkv, const float* __restrict__ cosT,
    const float* __restrict__ sinT) {
  int idx = blockIdx.x * 256 + threadIdx.x;   // 0 .. 4194303
  int t   = idx >> 10;                        // token 0..4095
  int rem = idx & 1023;
  int s   = rem >> 9;                         // 0 = q, 1 = k
  int rr  = rem & 511;
  int h   = rr >> 5;                          // head
  int p   = rr & 31;                          // pair index (HD/2 = 32)
  int l   = t & 2047;                         // position within sequence
  size_t base = (size_t)t * 3072 + s * 1024 + h * 64 + 2 * p;
  float c  = cosT[l * 32 + p];
  float sn = sinT[l * 32 + p];
  float x1 = qkv[base];
  float x2 = qkv[base + 1];
  qkv[base]     = x1 * c - x2 * sn;
  qkv[base + 1] = x1 * sn + x2 * c;
}

// ---------------------------------------------------------------------------
// Flash attention. Grid: (L/128, H, B). Block: 256 threads = 8 waves.
// Each wave owns 16 query rows; K/V 16x64 tiles double-buffered in LDS via
// async-to-LDS. Online softmax; P converted D-layout -> A-layout through a
// per-wave LDS patch, then P*V via WMMA.
// ---------------------------------------------------------------------------
__global__ __launch_bounds__(256) void attn_kernel(
    const float* __restrict__ qkv, const int* __restrict__ mask,
    float* __restrict__ out) {
  __shared__ __attribute__((aligned(16))) float Ks[2][16 * 64];
  __shared__ __attribute__((aligned(16))) float Vs[2][16 * 64];
  __shared__ __attribute__((aligned(16))) float Ps[8 * 256];  // 16x16 P tile per wave

  const int tid  = threadIdx.x;
  const int lane = tid & 31;
  const int wid  = tid >> 5;
  const int half = lane >> 4;
  const int l16  = lane & 15;
  const int h    = blockIdx.y;
  const int b    = blockIdx.z;
  const int q0   = blockIdx.x * 128 + wid * 16;
  const float scale = 0.125f;  // 64^-0.5, folded into Q

  const int krow = tid >> 4;
  const int kcv  = (tid & 15) * 4;
  auto issue_kv = [&](int buf, int kt) {
    size_t gb = (size_t)(b * 2048 + kt * 16 + krow) * 3072 + h * 64 + kcv;
    async_copy16(&Ks[buf][krow * 64 + kcv], qkv + gb + 1024);
    async_copy16(&Vs[buf][krow * 64 + kcv], qkv + gb + 2048);
  };

  // Q A-fragments for all 16 k-steps over HD=64, pre-scaled.
  v2f qa[16];
  {
    const float* qrow = qkv + (size_t)(b * 2048 + q0 + l16) * 3072 + h * 64;
#pragma unroll
    for (int es = 0; es < 16; ++es) {
      v2f t2 = *(const v2f*)(qrow + es * 4 + 2 * half);
      qa[es].x = t2.x * scale;
      qa[es].y = t2.y * scale;
    }
  }

  float m_st[8], l_st[8];
  v8f o[4] = {};
#pragma unroll
  for (int r = 0; r < 8; ++r) { m_st[r] = -1e30f; l_st[r] = 0.0f; }
  float* myP = Ps + wid * 256;

  issue_kv(0, 0);
  for (int kt = 0; kt < 128; ++kt) {
    const int cur = kt & 1;
    const int k0  = kt * 16;
    wait_async0();
    __syncthreads();
    if (kt + 1 < 128) issue_kv(cur ^ 1, kt + 1);

    const float* ks = Ks[cur];
    const float* vs = Vs[cur];

    // S = (Q*scale) . K^T : 16 WMMA k-steps over HD=64.
    v8f s = {};
#pragma unroll
    for (int es = 0; es < 16; ++es) {
      // B frag: B[kk][n] = K[n][es*4 + kk]; lane holds key-col n = l16.
      v2f kb = *(const v2f*)(&ks[l16 * 64 + es * 4 + 2 * half]);
      s = WMMA_F32(qa[es], kb, s);
    }

    // Additive mask depends only on key index (= lane&15 within tile).
    float madd = mask[b * 2048 + k0 + l16] ? 0.0f : -10000.0f;
#pragma unroll
    for (int r = 0; r < 8; ++r) s[r] += madd;

    // Online softmax. Row of VGPR r is r (lanes 0-15) or r+8 (lanes 16-31);
    // reductions stay within each 16-lane half via width-16 xor shuffles.
    float p[8];
#pragma unroll
    for (int r = 0; r < 8; ++r) {
      float v = s[r];
      for (int off = 8; off >= 1; off >>= 1) v = fmaxf(v, __shfl_xor(v, off, 16));
      float mn = fmaxf(m_st[r], v);
      float a  = __expf(m_st[r] - mn);
      float pv = __expf(s[r] - mn);
      float rs = pv;
      for (int off = 8; off >= 1; off >>= 1) rs += __shfl_xor(rs, off, 16);
      l_st[r] = l_st[r] * a + rs;
      m_st[r] = mn;
      p[r]    = pv;
#pragma unroll
      for (int t = 0; t < 4; ++t) o[t][r] *= a;
    }

    // P: D-layout -> per-wave LDS patch (row-major 16x16) -> A-layout frags.
#pragma unroll
    for (int r = 0; r < 8; ++r) myP[(r + 8 * half) * 16 + l16] = p[r];

#pragma unroll
    for (int et = 0; et < 4; ++et) {
#pragma unroll
      for (int kss = 0; kss < 4; ++kss) {
        v2f pa = *(const v2f*)(&myP[l16 * 16 + kss * 4 + 2 * half]);
        v2f vb;
        vb.x = vs[(kss * 4 + 2 * half) * 64 + et * 16 + l16];
        vb.y = vs[(kss * 4 + 2 * half + 1) * 64 + et * 16 + l16];
        o[et] = WMMA_F32(pa, vb, o[et]);
      }
    }
    __syncthreads();  // reads of buffer [cur] done before refill at kt+2
  }

  // Normalize and write attn output in [b, l, h*64+e] layout (GEMM-ready).
#pragma unroll
  for (int et = 0; et < 4; ++et)
#pragma unroll
    for (int r = 0; r < 8; ++r) {
      int row = q0 + r + 8 * half;
      int col = h * 64 + et * 16 + l16;
      out[(size_t)(b * 2048 + row) * 1024 + col] = o[et][r] / l_st[r];
    }
}

// ---------------------------------------------------------------------------
// Launch: qkv GEMM -> RoPE -> flash attention -> projection GEMM.
// d_ws layout: [0, 48MB) qkv (4096x3072 f32), [48MB, 64MB) attn out (4096x1024).
// ---------------------------------------------------------------------------
extern "C" void kernel_launch(void* const* d_in, const int* in_sizes, int n_in,
                              void* d_out, int out_size, void* d_ws, size_t ws_size,
                              hipStream_t stream) {
  const float* x     = (const float*)d_in[0];
  const int*   mask  = (const int*)d_in[1];
  const float* rc    = (const float*)d_in[2];
  const float* rs    = (const float*)d_in[3];
  const float* Wqkv  = (const float*)d_in[4];
  const float* Wproj = (const float*)d_in[5];
  float* out  = (float*)d_out;
  float* qkv  = (float*)d_ws;                    // 4096*3072 floats
  float* attn = qkv + (size_t)4096 * 3072;       // 4096*1024 floats

  // qkv = x @ W_qkv : M=4096, N=3072, K=1024
  gemm_f32_kernel<<<dim3(3072 / 64, 4096 / 128), 256, 0, stream>>>(
      x, Wqkv, qkv, 4096, 3072, 1024);
  // RoPE in place on q,k thirds: 4096*2*512 pairs
  rope_kernel<<<(4096 * 1024) / 256, 256, 0, stream>>>(qkv, rc, rs);
  // attention
  attn_kernel<<<dim3(2048 / 128, 16, 2), 256, 0, stream>>>(qkv, mask, attn);
  // out = attn @ W_proj : M=4096, N=1024, K=1024
  gemm_f32_kernel<<<dim3(1024 / 64, 4096 / 128), 256, 0, stream>>>(
      attn, Wproj, out, 4096, 1024, 1024);
}